// Symbolic_KANLayer_78005196030432
// MI455X (gfx1250) — compile-verified
//
#include <hip/hip_runtime.h>

typedef float v2f __attribute__((ext_vector_type(2)));
typedef float v4f __attribute__((ext_vector_type(4)));
typedef float v8f __attribute__((ext_vector_type(8)));

#define BATCH   4096
#define IN_DIM  128
#define OUT_DIM 128
#define BT      16      // batch tile per block (and WMMA M)

__global__ __launch_bounds__(256)
void kan_sin_kernel(const float* __restrict__ x,
                    const float* __restrict__ affine,
                    const float* __restrict__ mask,
                    float* __restrict__ y,
                    float* __restrict__ postacts)
{
    __shared__ float lds_x[BT * IN_DIM];        // 8 KB: x tile [b][i]
    __shared__ float lds_part[8][BT][32];       // 16 KB: per-wave partials [b][lane]

    const int tid  = threadIdx.x;
    const int lane = tid & 31;
    const int wave = tid >> 5;
    const int b0   = blockIdx.x * BT;           // batch tile base
    const int jb   = blockIdx.y * 16;           // j group base (8 waves x 2 iters)

    // ---- stage x tile into LDS (coalesced b128) ----
    const v4f* xg  = (const v4f*)(x + (size_t)b0 * IN_DIM);
    v4f*       lx4 = (v4f*)lds_x;
    #pragma unroll
    for (int k = tid; k < (BT * IN_DIM) / 4; k += 256)
        lx4[k] = xg[k];
    __syncthreads();

    for (int jj = 0; jj < 2; ++jj) {
        const int j = jb + wave + jj * 8;

        // lane handles i = 4*lane .. 4*lane+3
        const v4f* af = (const v4f*)(affine + ((size_t)j * IN_DIM + 4 * lane) * 4);
        v4f A0 = af[0], A1 = af[1], A2 = af[2], A3 = af[3];   // [a,b,c,d] per i
        v4f mk = *(const v4f*)(mask + (size_t)j * IN_DIM + 4 * lane);

        float a_[4] = {A0.x, A1.x, A2.x, A3.x};
        float bb[4] = {A0.y, A1.y, A2.y, A3.y};
        float mc[4] = {mk.x * A0.z, mk.y * A1.z, mk.z * A2.z, mk.w * A3.z};
        float md[4] = {mk.x * A0.w, mk.y * A1.w, mk.z * A2.w, mk.w * A3.w};

        float acc[BT];
        #pragma unroll
        for (int b = 0; b < BT; ++b) {
            v4f xv = *(const v4f*)(&lds_x[b * IN_DIM + 4 * lane]);
            float p0 = fmaf(mc[0], __sinf(fmaf(a_[0], xv.x, bb[0])), md[0]);
            float p1 = fmaf(mc[1], __sinf(fmaf(a_[1], xv.y, bb[1])), md[1]);
            float p2 = fmaf(mc[2], __sinf(fmaf(a_[2], xv.z, bb[2])), md[2]);
            float p3 = fmaf(mc[3], __sinf(fmaf(a_[3], xv.w, bb[3])), md[3]);
            v4f pv = {p0, p1, p2, p3};
            // write-only 256MB stream: non-temporal, coalesced 512B per wave-store
            v4f* pdst = (v4f*)(postacts +
                ((size_t)(b0 + b) * OUT_DIM + j) * IN_DIM + 4 * lane);
            __builtin_nontemporal_store(pv, pdst);
            acc[b] = (p0 + p1) + (p2 + p3);
        }

        // ---- transpose per-lane partials through LDS ----
        #pragma unroll
        for (int b = 0; b < BT; ++b)
            lds_part[wave][b][lane] = acc[b];
        __syncthreads();

        // ---- y[b] = sum over 32 lanes, via V_WMMA_F32_16X16X4_F32 against ones ----
        // A layout (16x4 f32): lane l holds A[l&15][2*(l>>4)+{0,1}] in its 2 VGPRs
        v8f c8 = {0.f,0.f,0.f,0.f,0.f,0.f,0.f,0.f};
        v2f ones = {1.0f, 1.0f};
        const float* pw  = &lds_part[wave][0][0];
        const int    row = (lane & 15) * 32 + 2 * (lane >> 4);
        #pragma unroll
        for (int c = 0; c < 8; ++c) {
            v2f a2 = *(const v2f*)(pw + row + 4 * c);
            c8 = __builtin_amdgcn_wmma_f32_16x16x4_f32(
                    /*neg_a=*/false, a2, /*neg_b=*/false, ones,
                    /*c_mod=*/(short)0, c8, /*reuse_a=*/false, /*reuse_b=*/false);
        }

        // D columns are identical; lane 0 holds y[M=0..7], lane 16 holds y[M=8..15]
        if ((lane & 15) == 0) {
            const int mbase = (lane >> 4) * 8;
            #pragma unroll
            for (int v = 0; v < 8; ++v)
                __builtin_nontemporal_store(c8[v],
                    y + (size_t)(b0 + mbase + v) * OUT_DIM + j);
        }
        __syncthreads();
    }
}

extern "C" void kernel_launch(void* const* d_in, const int* in_sizes, int n_in,
                              void* d_out, int out_size, void* d_ws, size_t ws_size,
                              hipStream_t stream) {
    (void)in_sizes; (void)n_in; (void)d_ws; (void)ws_size; (void)out_size;
    const float* x      = (const float*)d_in[0];
    const float* affine = (const float*)d_in[1];
    const float* mask   = (const float*)d_in[2];
    float* y        = (float*)d_out;                          // 4096*128
    float* postacts = y + (size_t)BATCH * OUT_DIM;            // 4096*128*128
    dim3 grid(BATCH / BT, OUT_DIM / 16);
    kan_sin_kernel<<<grid, 256, 0, stream>>>(x, affine, mask, y, postacts);
}